// Merge_Model_61735859912841
// MI455X (gfx1250) — compile-verified
//
#include <hip/hip_runtime.h>

// Problem constants (from the reference)
#define DDIM 300      // embed dim
#define PDIM 320      // padded dim: multiple of 64 (block N tile) and 4 (WMMA K step)
#define NWN  20000    // word-graph dst nodes
#define EWN  320000   // word-graph edges per relation
#define BN   256      // doc batch
#define LN   256      // seq length
#define SDN  2048     // doc-graph src nodes
#define EDN  4096     // doc-graph edges per relation
#define CN   20       // classes
#define KC   160      // K chunk staged in LDS (2 chunks for K=320)

typedef __attribute__((ext_vector_type(2))) float v2f;
typedef __attribute__((ext_vector_type(8))) float v8f;

// ---------------- utility kernels ----------------

__global__ void zero_f32(float* __restrict__ p, long long n) {
    long long t = (long long)blockIdx.x * blockDim.x + threadIdx.x;
    if (t < n) p[t] = 0.f;
}

// out[n, PDIM] = emb[nid[n], 0:300] zero-padded to PDIM cols
__global__ void gather_pad(const float* __restrict__ emb, const int* __restrict__ nid,
                           float* __restrict__ out, int n) {
    int t = blockIdx.x * blockDim.x + threadIdx.x;
    if (t >= n * PDIM) return;
    int r = t / PDIM, j = t % PDIM;
    out[t] = (j < DDIM) ? emb[(size_t)nid[r] * DDIM + j] : 0.f;
}

// pack [600,300] SAGE weight -> two [PDIM,PDIM] zero-padded halves (top=W[:300], bot=W[300:])
__global__ void pack_sage_w(const float* __restrict__ w, float* __restrict__ out) {
    int t = blockIdx.x * blockDim.x + threadIdx.x;
    if (t >= 2 * PDIM * PDIM) return;
    int h = t / (PDIM * PDIM);
    int r = (t / PDIM) % PDIM;
    int c = t % PDIM;
    out[t] = (r < DDIM && c < DDIM) ? w[((size_t)(h * DDIM + r)) * DDIM + c] : 0.f;
}

// pack [300,300] -> [PDIM,PDIM] zero-padded
__global__ void pack_w(const float* __restrict__ w, float* __restrict__ out) {
    int t = blockIdx.x * blockDim.x + threadIdx.x;
    if (t >= PDIM * PDIM) return;
    int r = t / PDIM, c = t % PDIM;
    out[t] = (r < DDIM && c < DDIM) ? w[(size_t)r * DDIM + c] : 0.f;
}

// ---------------- seg_mean (scatter + normalize) ----------------

__global__ void count_edges(const int* __restrict__ e_dst, int E, float* __restrict__ cnt) {
    int t = blockIdx.x * blockDim.x + threadIdx.x;
    if (t < E) atomicAdd(&cnt[e_dst[t]], 1.f);
}

// one WAVE per edge: edge indices are wave-uniform (scalarized), lanes stride the feature dim
__global__ void scatter_edges(const float* __restrict__ emb, const int* __restrict__ src_nid,
                              const int* __restrict__ e_src, const int* __restrict__ e_dst,
                              int E, float* __restrict__ sum) {
    const int lane = threadIdx.x & 31;
    const int e = blockIdx.x * (blockDim.x >> 5) + (threadIdx.x >> 5);
    if (e >= E) return;
    const int s = src_nid[e_src[e]];
    const float* __restrict__ src = emb + (size_t)s * DDIM;
    float* __restrict__ dst = sum + (size_t)e_dst[e] * PDIM;
    for (int j = lane; j < DDIM; j += 32)
        atomicAdd(&dst[j], src[j]);
}

__global__ void normalize_mean(float* __restrict__ sum, const float* __restrict__ cnt, int n) {
    int t = blockIdx.x * blockDim.x + threadIdx.x;
    if (t >= n * PDIM) return;
    sum[t] /= fmaxf(cnt[t / PDIM], 1.f);
}

// ---------------- fp32 WMMA GEMM, LDS-staged B ----------------
// C[M,N] = A[M,K] @ B[K,N] + (accum ? C : bias). N%64==0, K%KC==0, KC%4==0, pads zero.
// Block = 8 waves: block tile 128(M) x 64(N); wave computes 16 x 64 (4 WMMA tiles).
// B panel (KC x 64) staged to LDS transposed -> each B fragment is one ds_load_b64.
// Per 4 v_wmma: 1 global b64 (A) + 4 ds b64 (B).  WMMA only under wave-uniform branch.
__global__ __launch_bounds__(256) void gemm_f32_wmma(
        const float* __restrict__ A, const float* __restrict__ B,
        const float* __restrict__ bias, float* __restrict__ C,
        int M, int N, int K, int lda, int ldb, int ldc, int accum) {
    __shared__ float ldsB[64 * KC];          // 40 KB: ldsB[c*KC + k]  (B transposed)
    const int lane = threadIdx.x & 31;
    const int wave = threadIdx.x >> 5;
    const int hi   = lane >> 4;              // half-wave: K pair select
    const int l16  = lane & 15;
    const int nbase = blockIdx.y * 64;
    const int mt    = blockIdx.x * 8 + wave; // this wave's 16-row tile
    const bool valid = (mt * 16) < M;
    const int m = mt * 16 + l16;

    v8f acc[4];
    if (valid) {
        if (accum) {
#pragma unroll
            for (int ns = 0; ns < 4; ++ns)
#pragma unroll
                for (int i = 0; i < 8; ++i)
                    acc[ns][i] = C[(size_t)(mt * 16 + i + hi * 8) * ldc + nbase + ns * 16 + l16];
        } else {
#pragma unroll
            for (int ns = 0; ns < 4; ++ns) {
                const int n = nbase + ns * 16 + l16;
                const float bv = (bias && n < DDIM) ? bias[n] : 0.f;
#pragma unroll
                for (int i = 0; i < 8; ++i) acc[ns][i] = bv;
            }
        }
    }

    const float* __restrict__ Arow = A + (size_t)m * lda;
    for (int kk = 0; kk < K; kk += KC) {
        __syncthreads();                      // all waves (incl. invalid) participate
        // cooperative stage: consecutive threads -> consecutive global n (coalesced)
        for (int idx = threadIdx.x; idx < 64 * KC; idx += 256) {
            int c = idx & 63, kloc = idx >> 6;
            ldsB[c * KC + kloc] = B[(size_t)(kk + kloc) * ldb + nbase + c];
        }
        __syncthreads();
        if (valid) {
            for (int k = 0; k < KC; k += 4) {
                v2f a = *(const v2f*)(Arow + kk + k + 2 * hi);   // A[m][k+2h .. +1]
#pragma unroll
                for (int ns = 0; ns < 4; ++ns) {
                    v2f b = *(const v2f*)(&ldsB[(ns * 16 + l16) * KC + k + 2 * hi]);
                    acc[ns] = __builtin_amdgcn_wmma_f32_16x16x4_f32(
                        false, a, false, b, (short)0, acc[ns], false, false);
                }
            }
        }
    }
    if (valid) {
#pragma unroll
        for (int ns = 0; ns < 4; ++ns)
#pragma unroll
            for (int i = 0; i < 8; ++i)
                C[(size_t)(mt * 16 + i + hi * 8) * ldc + nbase + ns * 16 + l16] = acc[ns][i];
    }
}

// ---------------- sdpa_fuse (+ residual add) ----------------
// One wave per node. mean_t distributes: fused = sum_s w_s * X_s, w_s = mean_t softmax(S*scale)[t,s]
__global__ void sdpa_fuse_add(const float* __restrict__ h0, const float* __restrict__ h1,
                              const float* __restrict__ h2, const float* __restrict__ addv,
                              float* __restrict__ out, int n) {
    const int lane = threadIdx.x & 31;
    const int node = blockIdx.x * (blockDim.x >> 5) + (threadIdx.x >> 5);
    if (node >= n) return;
    const float* x0 = h0 + (size_t)node * PDIM;
    const float* x1 = h1 + (size_t)node * PDIM;
    const float* x2 = h2 + (size_t)node * PDIM;

    float d00 = 0, d01 = 0, d02 = 0, d11 = 0, d12 = 0, d22 = 0;
    for (int j = lane; j < PDIM; j += 32) {   // pad cols are zero -> harmless
        float a = x0[j], b = x1[j], e = x2[j];
        d00 += a * a; d01 += a * b; d02 += a * e;
        d11 += b * b; d12 += b * e; d22 += e * e;
    }
    for (int off = 16; off > 0; off >>= 1) {
        d00 += __shfl_xor(d00, off, 32);
        d01 += __shfl_xor(d01, off, 32);
        d02 += __shfl_xor(d02, off, 32);
        d11 += __shfl_xor(d11, off, 32);
        d12 += __shfl_xor(d12, off, 32);
        d22 += __shfl_xor(d22, off, 32);
    }
    const float scale = rsqrtf((float)DDIM);  // x.shape[-1] == 300
    float S[3][3] = {{d00, d01, d02}, {d01, d11, d12}, {d02, d12, d22}};
    float wsum[3] = {0.f, 0.f, 0.f};
    for (int t = 0; t < 3; ++t) {
        float p0 = S[t][0] * scale, p1 = S[t][1] * scale, p2 = S[t][2] * scale;
        float mx = fmaxf(p0, fmaxf(p1, p2));
        float e0 = __expf(p0 - mx), e1 = __expf(p1 - mx), e2 = __expf(p2 - mx);
        float inv = 1.f / (e0 + e1 + e2);
        wsum[0] += e0 * inv; wsum[1] += e1 * inv; wsum[2] += e2 * inv;
    }
    const float w0 = wsum[0] * (1.f / 3.f), w1 = wsum[1] * (1.f / 3.f), w2 = wsum[2] * (1.f / 3.f);
    const float* av = addv + (size_t)node * PDIM;
    float* o = out + (size_t)node * PDIM;
    for (int j = lane; j < PDIM; j += 32)
        o[j] = w0 * x0[j] + w1 * x1[j] + w2 * x2[j] + av[j];
}

// ---------------- sequence mean pool (exploits linearity of the dense layer) ----------------
__global__ void pool_seq(const float* __restrict__ table, const int* __restrict__ xb,
                         float* __restrict__ pooled) {
    const int b = blockIdx.x;
    const int j = threadIdx.x;
    if (j >= PDIM) return;
    const int* row = xb + b * LN;
    float acc = 0.f;
    for (int l = 0; l < LN; ++l) {
        int idx = row[l];
        if (idx < NWN) acc += table[(size_t)idx * PDIM + j];  // idx==NWN -> zero row
    }
    pooled[(size_t)b * PDIM + j] = acc * (1.f / (float)LN);
}

// ---------------- final classifier (3 MFLOP: scalar kernel is cheapest) ----------------
__global__ void fc_kernel(const float* __restrict__ rep, const float* __restrict__ wfc,
                          const float* __restrict__ bfc, float* __restrict__ out) {
    int t = blockIdx.x * blockDim.x + threadIdx.x;
    if (t >= BN * CN) return;
    int b = t / CN, c = t % CN;
    float acc = bfc[c];
    const float* r = rep + (size_t)b * PDIM;
    for (int j = 0; j < DDIM; ++j) acc += r[j] * wfc[(size_t)j * CN + c];
    out[t] = acc;
}

// ---------------- host orchestration ----------------

extern "C" void kernel_launch(void* const* d_in, const int* in_sizes, int n_in,
                              void* d_out, int out_size, void* d_ws, size_t ws_size,
                              hipStream_t stream) {
    (void)in_sizes; (void)n_in; (void)out_size; (void)ws_size;
    const float* emb_word = (const float*)d_in[0];
    const float* emb_doc  = (const float*)d_in[1];
    const float* w_sage[6] = {(const float*)d_in[2],  (const float*)d_in[4],  (const float*)d_in[6],
                              (const float*)d_in[8],  (const float*)d_in[10], (const float*)d_in[12]};
    const float* b_sage[6] = {(const float*)d_in[3],  (const float*)d_in[5],  (const float*)d_in[7],
                              (const float*)d_in[9],  (const float*)d_in[11], (const float*)d_in[13]};
    const float* w_dense = (const float*)d_in[14];
    const float* b_dense = (const float*)d_in[15];
    const float* w_fc    = (const float*)d_in[16];
    const float* b_fc    = (const float*)d_in[17];
    const int* dst_nid      = (const int*)d_in[18];
    const int* src_nid_w[3] = {(const int*)d_in[19], (const int*)d_in[20], (const int*)d_in[21]};
    const int* e_src_w[3]   = {(const int*)d_in[22], (const int*)d_in[24], (const int*)d_in[26]};
    const int* e_dst_w[3]   = {(const int*)d_in[23], (const int*)d_in[25], (const int*)d_in[27]};
    const int* x_batch      = (const int*)d_in[28];
    const int* src_nid_d[3] = {(const int*)d_in[30], (const int*)d_in[31], (const int*)d_in[32]};
    const int* e_src_d[3]   = {(const int*)d_in[33], (const int*)d_in[35], (const int*)d_in[37]};
    const int* e_dst_d[3]   = {(const int*)d_in[34], (const int*)d_in[36], (const int*)d_in[38]};
    float* out = (float*)d_out;

    // workspace carve-up (all fp32, PDIM-padded)
    float* ws = (float*)d_ws;
    size_t off = 0;
    auto carve = [&](size_t nflt) { float* p = ws + off; off += nflt; return p; };
    float* hdst = carve((size_t)NWN * PDIM);           // emb_word[dst_nid], padded
    float* nbuf = carve((size_t)NWN * PDIM);           // neigh scratch, later = word table
    float* h3[3]; for (int r = 0; r < 3; ++r) h3[r] = carve((size_t)NWN * PDIM);
    float* cnt  = carve(NWN);                           // edge counts (reused for docs)
    float* wpk[6]; for (int r = 0; r < 6; ++r) wpk[r] = carve((size_t)2 * PDIM * PDIM);
    float* wdn  = carve((size_t)PDIM * PDIM);
    float* pooled = carve((size_t)BN * PDIM);
    float* docout = carve((size_t)BN * PDIM);
    float* dnb    = carve((size_t)BN * PDIM);
    float* d3[3]; for (int r = 0; r < 3; ++r) d3[r] = carve((size_t)BN * PDIM);
    float* docrep = carve((size_t)BN * PDIM);

    const int TB = 256;
    const unsigned wordElems = (unsigned)(((size_t)NWN * PDIM + TB - 1) / TB);
    const unsigned docElems  = (unsigned)(((size_t)BN * PDIM + TB - 1) / TB);
    const dim3 gemmGridW((NWN + 127) / 128, PDIM / 64);   // 157 x 5
    const dim3 gemmGridD((BN + 127) / 128, PDIM / 64);    // 2 x 5

    // pack weights once per call
    for (int r = 0; r < 6; ++r)
        pack_sage_w<<<(2 * PDIM * PDIM + TB - 1) / TB, TB, 0, stream>>>(w_sage[r], wpk[r]);
    pack_w<<<(PDIM * PDIM + TB - 1) / TB, TB, 0, stream>>>(w_dense, wdn);

    // h_dst = emb_word[dst_nid]
    gather_pad<<<wordElems, TB, 0, stream>>>(emb_word, dst_nid, hdst, NWN);

    // word-graph SAGE per relation: h_r = hdst @ W_top + b + seg_mean @ W_bot
    for (int r = 0; r < 3; ++r) {
        zero_f32<<<wordElems, TB, 0, stream>>>(nbuf, (long long)NWN * PDIM);
        zero_f32<<<(NWN + TB - 1) / TB, TB, 0, stream>>>(cnt, NWN);
        count_edges<<<(EWN + TB - 1) / TB, TB, 0, stream>>>(e_dst_w[r], EWN, cnt);
        scatter_edges<<<(EWN + 7) / 8, TB, 0, stream>>>(emb_word, src_nid_w[r], e_src_w[r],
                                                        e_dst_w[r], EWN, nbuf);
        normalize_mean<<<wordElems, TB, 0, stream>>>(nbuf, cnt, NWN);
        gemm_f32_wmma<<<gemmGridW, TB, 0, stream>>>(hdst, wpk[r], b_sage[r], h3[r],
                                                    NWN, PDIM, PDIM, PDIM, PDIM, PDIM, 0);
        gemm_f32_wmma<<<gemmGridW, TB, 0, stream>>>(nbuf, wpk[r] + (size_t)PDIM * PDIM, nullptr,
                                                    h3[r], NWN, PDIM, PDIM, PDIM, PDIM, PDIM, 1);
    }

    // attention fuse + residual -> word table (reuse nbuf)
    sdpa_fuse_add<<<(NWN + 7) / 8, 256, 0, stream>>>(h3[0], h3[1], h3[2], hdst, nbuf, NWN);

    // mean-pool sequences (linearity: pool before dense), then dense GEMM
    pool_seq<<<BN, PDIM, 0, stream>>>(nbuf, x_batch, pooled);
    gemm_f32_wmma<<<gemmGridD, TB, 0, stream>>>(pooled, wdn, b_dense, docout,
                                                BN, PDIM, PDIM, PDIM, PDIM, PDIM, 0);

    // doc-graph SAGE per relation
    for (int r = 0; r < 3; ++r) {
        zero_f32<<<docElems, TB, 0, stream>>>(dnb, (long long)BN * PDIM);
        zero_f32<<<1, TB, 0, stream>>>(cnt, BN);
        count_edges<<<(EDN + TB - 1) / TB, TB, 0, stream>>>(e_dst_d[r], EDN, cnt);
        scatter_edges<<<(EDN + 7) / 8, TB, 0, stream>>>(emb_doc, src_nid_d[r], e_src_d[r],
                                                        e_dst_d[r], EDN, dnb);
        normalize_mean<<<docElems, TB, 0, stream>>>(dnb, cnt, BN);
        gemm_f32_wmma<<<gemmGridD, TB, 0, stream>>>(docout, wpk[3 + r], b_sage[3 + r], d3[r],
                                                    BN, PDIM, PDIM, PDIM, PDIM, PDIM, 0);
        gemm_f32_wmma<<<gemmGridD, TB, 0, stream>>>(dnb, wpk[3 + r] + (size_t)PDIM * PDIM, nullptr,
                                                    d3[r], BN, PDIM, PDIM, PDIM, PDIM, PDIM, 1);
    }

    // doc fuse + residual, then classifier
    sdpa_fuse_add<<<(BN + 7) / 8, 256, 0, stream>>>(d3[0], d3[1], d3[2], docout, docrep, BN);
    fc_kernel<<<(BN * CN + TB - 1) / TB, TB, 0, stream>>>(docrep, w_fc, b_fc, out);
}